// MultiHeadGraphAttention_35682588295403
// MI455X (gfx1250) — compile-verified
//
#include <hip/hip_runtime.h>

// MultiHeadGraphAttention for MI455X (gfx1250, wave32)
//
// Phases (all on `stream`):
//  1) k_coeff   : C16[16][128] = w[h] (*) attn-halves, cols 8..15 zero-padded
//  2) k_zero    : deg[N] = 0
//  3) k_degree  : deg[src]++ per edge (int atomics)
//  4) k_scan    : exclusive scan -> offsets[N+1], cursor[N]   (single 1024-thread block)
//  5) k_fill    : CSR column fill via cursor atomics
//  6) k_pscore  : P[N][8] = x @ C^T via V_WMMA_F32_16X16X4_F32 (one wave per 16-row tile)
//  7) k_gather  : one wave32 per node; atomic-free accumulation of 4 heads x 128 dims,
//                 normalize by rowsum, scale by w, write out[H][N][128].

typedef __attribute__((ext_vector_type(2))) float v2f;
typedef __attribute__((ext_vector_type(8))) float v8f;

#define DD 128
#define HH 4

// ---------------------------------------------------------------- coeff matrix
// C16[col][k]: col 0..3 = head h src-half (w[h][k]*attn[h][k]),
//              col 4..7 = head h dst-half (w[h][k]*attn[h][128+k]),
//              col 8..15 = 0 (padding so WMMA B loads are unconditional)
__global__ void k_coeff(const float* __restrict__ w, const float* __restrict__ attn,
                        float* __restrict__ C16) {
  int t = blockIdx.x * blockDim.x + threadIdx.x;
  if (t >= 16 * DD) return;
  int col = t / DD, k = t % DD;
  float v = 0.0f;
  if (col < 8) {
    int h = col & 3;
    float a = (col < 4) ? attn[h * 2 * DD + k] : attn[h * 2 * DD + DD + k];
    v = w[h * DD + k] * a;
  }
  C16[col * DD + k] = v;
}

__global__ void k_zero_i32(int* __restrict__ p, int n) {
  int i = blockIdx.x * blockDim.x + threadIdx.x;
  if (i < n) p[i] = 0;
}

__global__ void k_degree(const int* __restrict__ edge, int E, int* __restrict__ deg) {
  for (int e = blockIdx.x * blockDim.x + threadIdx.x; e < E; e += gridDim.x * blockDim.x)
    atomicAdd(&deg[edge[e]], 1);
}

// single-block exclusive scan (Hillis-Steele per 1024-chunk + running carry)
__global__ void k_scan(const int* __restrict__ deg, int N,
                       int* __restrict__ offsets, int* __restrict__ cursor) {
  __shared__ int lds[1024];
  int tid = threadIdx.x;
  int running = 0;
  for (int base = 0; base < N; base += 1024) {
    int i = base + tid;
    int v = (i < N) ? deg[i] : 0;
    lds[tid] = v;
    __syncthreads();
    for (int o = 1; o < 1024; o <<= 1) {
      int t2 = (tid >= o) ? lds[tid - o] : 0;
      __syncthreads();
      lds[tid] += t2;
      __syncthreads();
    }
    int incl = lds[tid];
    int total = lds[1023];
    if (i < N) {
      int ex = running + incl - v;
      offsets[i] = ex;
      cursor[i] = ex;
    }
    running += total;
    __syncthreads();
  }
  if (tid == 0) offsets[N] = running;
}

__global__ void k_fill(const int* __restrict__ edge, int E,
                       int* __restrict__ cursor, int* __restrict__ col) {
  for (int e = blockIdx.x * blockDim.x + threadIdx.x; e < E; e += gridDim.x * blockDim.x) {
    int s = edge[e];
    int d = edge[E + e];
    int pos = atomicAdd(&cursor[s], 1);
    col[pos] = d;
  }
}

// ---------------------------------------------------------------- WMMA GEMV
// P[n][0..7] = x[n][:] . C16[col][:]   via D = A(16x4) x B(4x16) + C, K-loop of 32.
// A layout (32b, 16x4): lanes 0-15 row M=lane hold K=0,1 in VGPR0,1; lanes 16-31 K=2,3.
// B layout mirrors: lane col N=lane&15, VGPR0,1 = K rows (koff, koff+1).
// C/D (32b, 16x16): VGPR r = row r (lanes 0-15) / row 8+r (lanes 16-31), N = lane&15.
__global__ void __launch_bounds__(256) k_pscore_wmma(
    const float* __restrict__ x, const float* __restrict__ C16,
    float* __restrict__ P, int ntiles, int N) {
  int wave = (blockIdx.x * blockDim.x + threadIdx.x) >> 5;
  int lane = threadIdx.x & 31;
  if (wave >= ntiles) return;
  int m = lane & 15;
  int half = lane >> 4;
  int koff = half * 2;
  int row = wave * 16 + m;
  const float* arow = x + (size_t)(row < N ? row : N - 1) * DD + koff;
  const float* brow = C16 + m * DD + koff;   // zero-padded: no predication needed
  v8f acc = {};
#pragma unroll 8
  for (int kk = 0; kk < DD; kk += 4) {
    v2f a = *(const v2f*)(arow + kk);   // global_load_b64
    v2f b = *(const v2f*)(brow + kk);   // global_load_b64
    acc = __builtin_amdgcn_wmma_f32_16x16x4_f32(
        /*neg_a=*/false, a, /*neg_b=*/false, b,
        /*c_mod=*/(short)0, acc, /*reuse_a=*/false, /*reuse_b=*/false);
  }
  int row0 = wave * 16 + (half ? 8 : 0);
  if (m < 8) {
#pragma unroll
    for (int r = 0; r < 8; ++r) {
      int rr = row0 + r;
      if (rr < N) P[(size_t)rr * 8 + m] = acc[r];
    }
  }
}

// ---------------------------------------------------------------- gather
__global__ void __launch_bounds__(256) k_gather(
    const float* __restrict__ x, const float* __restrict__ P,
    const float* __restrict__ w,
    const int* __restrict__ offsets, const int* __restrict__ col,
    float* __restrict__ out, int N) {
  int wave = (blockIdx.x * blockDim.x + threadIdx.x) >> 5;
  int lane = threadIdx.x & 31;
  if (wave >= N) return;
  const int i = wave;

  const float4 ps = *(const float4*)(P + (size_t)i * 8);   // src-half scores, one b128

  float4 acc[HH];
  float rs[HH];
#pragma unroll
  for (int h = 0; h < HH; ++h) {
    acc[h] = make_float4(0.f, 0.f, 0.f, 0.f);
    rs[h] = 0.f;
  }

  const int beg = offsets[i], end = offsets[i + 1];
  for (int p = beg; p < end; ++p) {
    int j = col[p];
    if (p + 1 < end) {
      int jn = col[p + 1];
      __builtin_prefetch(x + (size_t)jn * DD + lane * 4, 0, 0);  // global_prefetch_b8
    }
    const float4 xv = *(const float4*)(x + (size_t)j * DD + lane * 4);
    const float4 pd = *(const float4*)(P + (size_t)j * 8 + 4);   // dst-half, one b128
    float s0 = ps.x + pd.x, s1 = ps.y + pd.y, s2 = ps.z + pd.z, s3 = ps.w + pd.w;
    float e0 = __expf(-((s0 > 0.f) ? s0 : 0.2f * s0));
    float e1 = __expf(-((s1 > 0.f) ? s1 : 0.2f * s1));
    float e2 = __expf(-((s2 > 0.f) ? s2 : 0.2f * s2));
    float e3 = __expf(-((s3 > 0.f) ? s3 : 0.2f * s3));
    rs[0] += e0; rs[1] += e1; rs[2] += e2; rs[3] += e3;
    acc[0].x = fmaf(e0, xv.x, acc[0].x); acc[0].y = fmaf(e0, xv.y, acc[0].y);
    acc[0].z = fmaf(e0, xv.z, acc[0].z); acc[0].w = fmaf(e0, xv.w, acc[0].w);
    acc[1].x = fmaf(e1, xv.x, acc[1].x); acc[1].y = fmaf(e1, xv.y, acc[1].y);
    acc[1].z = fmaf(e1, xv.z, acc[1].z); acc[1].w = fmaf(e1, xv.w, acc[1].w);
    acc[2].x = fmaf(e2, xv.x, acc[2].x); acc[2].y = fmaf(e2, xv.y, acc[2].y);
    acc[2].z = fmaf(e2, xv.z, acc[2].z); acc[2].w = fmaf(e2, xv.w, acc[2].w);
    acc[3].x = fmaf(e3, xv.x, acc[3].x); acc[3].y = fmaf(e3, xv.y, acc[3].y);
    acc[3].z = fmaf(e3, xv.z, acc[3].z); acc[3].w = fmaf(e3, xv.w, acc[3].w);
  }

#pragma unroll
  for (int h = 0; h < HH; ++h) {
    float inv = 1.0f / rs[h];
    const float4 wv = *(const float4*)(w + h * DD + lane * 4);
    float4 o;
    o.x = acc[h].x * wv.x * inv;
    o.y = acc[h].y * wv.y * inv;
    o.z = acc[h].z * wv.z * inv;
    o.w = acc[h].w * wv.w * inv;
    *(float4*)(out + ((size_t)h * N + i) * DD + lane * 4) = o;
  }
}

// ---------------------------------------------------------------- launch
extern "C" void kernel_launch(void* const* d_in, const int* in_sizes, int n_in,
                              void* d_out, int out_size, void* d_ws, size_t ws_size,
                              hipStream_t stream) {
  const float* x    = (const float*)d_in[0];   // [N,128]
  const float* w    = (const float*)d_in[1];   // [4,1,128]
  const float* attn = (const float*)d_in[2];   // [4,256,1]
  const int*   edge = (const int*)d_in[3];     // [2,E]
  float* out = (float*)d_out;                  // [4,N,128]

  const int N = in_sizes[0] / DD;
  const int E = in_sizes[3] / 2;

  // workspace carve-up (256B-aligned regions)
  char* ws = (char*)d_ws;
  size_t off = 0;
  auto carve = [&](size_t bytes) -> void* {
    void* p = ws + off;
    off = (off + bytes + 255) & ~(size_t)255;
    return p;
  };
  float* C16    = (float*)carve(16 * DD * sizeof(float));
  float* P      = (float*)carve((size_t)N * 8 * sizeof(float));
  int* offsets  = (int*)carve((size_t)(N + 1) * sizeof(int));
  int* cursor   = (int*)carve((size_t)N * sizeof(int));
  int* deg      = (int*)carve((size_t)N * sizeof(int));
  int* colidx   = (int*)carve((size_t)E * sizeof(int));
  (void)ws_size;

  k_coeff<<<(16 * DD + 255) / 256, 256, 0, stream>>>(w, attn, C16);
  k_zero_i32<<<(N + 255) / 256, 256, 0, stream>>>(deg, N);
  k_degree<<<(E + 255) / 256, 256, 0, stream>>>(edge, E, deg);
  k_scan<<<1, 1024, 0, stream>>>(deg, N, offsets, cursor);
  k_fill<<<(E + 255) / 256, 256, 0, stream>>>(edge, E, cursor, colidx);

  const int ntiles = (N + 15) / 16;
  k_pscore_wmma<<<(ntiles + 7) / 8, 256, 0, stream>>>(x, C16, P, ntiles, N);
  k_gather<<<(N + 7) / 8, 256, 0, stream>>>(x, P, w, offsets, colidx, out, N);
}